// crps_loss_batch_mean_first_32469952758267
// MI455X (gfx1250) — compile-verified
//
#include <hip/hip_runtime.h>

// CRPS loss, fused single pass over fore[B,E,H,W] + obs[B,H,W], fp32.
// B=8, E=16, H=W=256.  Memory-bound design: each thread owns 4 consecutive
// pixels (float4 loads), streams 16 ensemble planes + obs (17x b128 loads),
// does ~1088 VALU ops, then a deterministic two-level reduction whose block
// and final stages use v_wmma_f32_16x16x4_f32 with an all-ones A matrix.

typedef __attribute__((ext_vector_type(2))) float v2f;
typedef __attribute__((ext_vector_type(8))) float v8f;

#define ENS    16
#define HW     65536        // 256*256
#define NBATCH 8
#define TOTAL  (NBATCH * HW)     // 524288 pixels
#define UNITS  (TOTAL / 4)       // 131072 float4 units
#define BLOCK  256
#define GRID   (UNITS / BLOCK)   // 512 blocks

// Sum 256 floats sitting in LDS using 4 chained WMMAs (one full wave, EXEC
// must be all-1s across the wave).  A = ones(16x4), B packs 64 values; since
// A is all ones, D row m = column sums of B, identical for every m, and
// sum_n D[0][n] = sum of all 64 B entries regardless of B's lane layout.
__device__ __forceinline__ float wave_wmma_sum256(const float* vals) {
  const int lane = threadIdx.x & 31;
  v2f a; a[0] = 1.0f; a[1] = 1.0f;
  v8f c = {};
#pragma unroll
  for (int k = 0; k < 4; ++k) {
    v2f b;
    b[0] = vals[k * 64 + lane];
    b[1] = vals[k * 64 + 32 + lane];
    c = __builtin_amdgcn_wmma_f32_16x16x4_f32(false, a, false, b,
                                              (short)0, c, false, false);
  }
  // c[0]: lanes 0-15 hold D[0][0..15], lanes 16-31 hold D[8][0..15] (equal).
  float s = c[0];
#pragma unroll
  for (int off = 16; off > 0; off >>= 1)
    s += __shfl_down(s, off, 32);
  return s * 0.5f;   // lane 0: total of the 256 inputs
}

__global__ void __launch_bounds__(BLOCK)
crps_main_kernel(const float* __restrict__ fore,
                 const float* __restrict__ obs,
                 float* __restrict__ block_partial) {
  __shared__ float lds[BLOCK];

  const int tid = threadIdx.x;
  const int u   = blockIdx.x * BLOCK + tid;   // float4 unit id
  const int b   = u >> 14;                    // / (HW/4)
  const int p   = (u & 16383) << 2;           // pixel offset within plane

  // obs: one float4
  const float4 o4 = *(const float4*)(obs + (size_t)b * HW + p);

  // fore: 16 coalesced b128 loads, one per ensemble plane
  float4 f4[ENS];
  const float* base = fore + ((size_t)b * ENS) * HW + p;
#pragma unroll
  for (int e = 0; e < ENS; ++e)
    f4[e] = *(const float4*)(base + (size_t)e * HW);

  float acc = 0.0f;
#pragma unroll
  for (int cmp = 0; cmp < 4; ++cmp) {
    float fv[ENS];
#pragma unroll
    for (int e = 0; e < ENS; ++e)
      fv[e] = (&f4[e].x)[cmp];
    const float ob = (&o4.x)[cmp];

    float s1 = 0.0f;            // sum |f_e - obs|
#pragma unroll
    for (int e = 0; e < ENS; ++e)
      s1 += fabsf(fv[e] - ob);

    float s2 = 0.0f;            // sum_{i<j} |f_i - f_j|
#pragma unroll
    for (int i = 0; i < ENS; ++i)
#pragma unroll
      for (int j = i + 1; j < ENS; ++j)
        s2 += fabsf(fv[i] - fv[j]);

    // term1 - 0.5*term2 = s1/16 - (2*s2/256)/2 = s1/16 - s2/256
    acc += s1 * 0.0625f - s2 * 0.00390625f;
  }

  lds[tid] = acc;
  __syncthreads();

  if (tid < 32) {                       // wave 0, full EXEC
    float s = wave_wmma_sum256(lds);
    if (tid == 0) block_partial[blockIdx.x] = s;
  }
}

__global__ void __launch_bounds__(32)
crps_final_kernel(const float* __restrict__ block_partial,
                  float* __restrict__ out) {
  const int lane = threadIdx.x;   // exactly one wave launched
  v2f a; a[0] = 1.0f; a[1] = 1.0f;
  v8f c = {};
#pragma unroll
  for (int k = 0; k < GRID / 64; ++k) {   // 512 partials, 8 chained WMMAs
    v2f b;
    b[0] = block_partial[k * 64 + lane];
    b[1] = block_partial[k * 64 + 32 + lane];
    c = __builtin_amdgcn_wmma_f32_16x16x4_f32(false, a, false, b,
                                              (short)0, c, false, false);
  }
  float s = c[0];
#pragma unroll
  for (int off = 16; off > 0; off >>= 1)
    s += __shfl_down(s, off, 32);
  if (lane == 0)
    out[0] = s * 0.5f * (1.0f / (float)TOTAL);  // global mean
}

extern "C" void kernel_launch(void* const* d_in, const int* in_sizes, int n_in,
                              void* d_out, int out_size, void* d_ws, size_t ws_size,
                              hipStream_t stream) {
  const float* fore = (const float*)d_in[0];   // [8,16,256,256] f32
  const float* obs  = (const float*)d_in[1];   // [8,256,256]    f32
  float* ws  = (float*)d_ws;                   // 512 block partials (2 KB)
  float* out = (float*)d_out;                  // 1 float

  crps_main_kernel<<<GRID, BLOCK, 0, stream>>>(fore, obs, ws);
  crps_final_kernel<<<1, 32, 0, stream>>>(ws, out);
}